// ReassemblePatchesLayer_42984032698840
// MI455X (gfx1250) — compile-verified
//
#include <hip/hip_runtime.h>
#include <math.h>

typedef __attribute__((ext_vector_type(2))) float v2f;
typedef __attribute__((ext_vector_type(4))) float v4f;
typedef __attribute__((ext_vector_type(8))) float v8f;

#define NB   128   // batch
#define NP    64   // patch size
#define NCH   16   // channels (gridsize^2)
#define MP   128   // padded canvas
#define TSZ  192   // mirrored 1-D table: index i <-> integer offset k = 95 - i
#define CST   68   // dword stride per column of sTr (pad for b64 align + banks)

// Real part of the closed-form Fourier-shift (Dirichlet) kernel for M=128:
// hr(k-d) = (1/128) cos(pi t/128) sin(pi t)/sin(pi t/128),  t = k - d,
// with sin(pi t) = -(-1)^k sin(pi d) (spd precomputed).
__device__ __forceinline__ float shift_hr(int k, float d, float spd) {
    const float PI_F = 3.14159265358979323846f;
    float t   = (float)k - d;
    float spt = (k & 1) ? spd : -spd;        // sin(pi * t)
    float bb  = t * (PI_F / 128.0f);
    float sb  = __sinf(bb);
    float cb  = __cosf(bb);
    float rsb = __builtin_amdgcn_rcpf(sb);   // v_rcp_f32 (no IEEE divide)
    float ratio = (__builtin_fabsf(sb) > 1e-7f) ? (spt * rsb) : 128.0f;
    return 0.0078125f * cb * ratio;
}

__global__ __launch_bounds__(256)
void reassemble_wmma(const float* __restrict__ patches,
                     const float* __restrict__ positions,
                     float* __restrict__ out) {
    __shared__ float sP [NP * NP];   // 16 KB: patch, row-major [u][v]
    __shared__ float sTr[64 * CST];  // 17 KB: T real, column-major [x][u], pad-68
    __shared__ float sHyR[TSZ];      // mirrored 1-D real shift tables (1.5 KB)
    __shared__ float sHxR[TSZ];
    __shared__ float sWS[8];         // per-wave partial checkerboard sums

    const int b    = blockIdx.x;
    const int tid  = threadIdx.x;
    const int w    = tid >> 5;       // wave id 0..7
    const int lane = tid & 31;
    const int lhi  = lane >> 4;      // 0 / 1 (lane half)
    const int ln   = lane & 15;      // 0..15
    const int koff = lhi * 2;        // K sub-offset inside a 16x16x4 fragment

    // Wave w owns output rows [16w, 16w+16) x all 128 cols: 8 f32 16x16 tiles.
    v8f acc[8];
#pragma unroll
    for (int i = 0; i < 8; ++i) acc[i] = (v8f){};

    float beta = 0.0f;   // accumulated rank-1 (imaginary-path) coefficient

    const float* pbase = patches + (size_t)b * NP * NP * NCH;

    for (int c = 0; c < NCH; ++c) {
        const float dy = positions[((size_t)b * NCH + c) * 2 + 0];
        const float dx = positions[((size_t)b * NCH + c) * 2 + 1];
        const float PI_F = 3.14159265358979323846f;
        const float spdy = __sinf(PI_F * dy);
        const float spdx = __sinf(PI_F * dx);

        // ---- load patch channel c into LDS; fold in checkerboard sum ----
        float psum = 0.0f;
        for (int i = tid; i < NP * NP; i += 256) {
            const float val = pbase[(size_t)i * NCH + c];
            sP[i] = val;
            // parity of (u + v) with u = i>>6, v = i&63
            psum += (((i >> 6) + i) & 1) ? -val : val;
            if (c + 1 < NCH)                      // hint next channel's line
                __builtin_prefetch(&pbase[(size_t)i * NCH + c + 1], 0, 1);
        }
        // wave-level reduce, then one slot per wave
#pragma unroll
        for (int m = 16; m > 0; m >>= 1) psum += __shfl_xor(psum, m, 32);
        if (lane == 0) sWS[w] = psum;

        // ---- fill MIRRORED real tables: table[i] = hr(95 - i) ----
        // Fragment index idx = (u|v) - (y|x) + 127 is ascending in j.
        for (int i = tid; i < 2 * TSZ; i += 256) {
            const int dir = i / TSZ;              // 0 = y table, 1 = x table
            const int idx = i - dir * TSZ;
            const int k   = 95 - idx;             // mirrored integer offset
            if (dir) sHxR[idx] = shift_hr(k, dx, spdx);
            else     sHyR[idx] = shift_hr(k, dy, spdy);
        }

        __syncthreads();  // sP + tables + per-wave sums ready

        // rank-1 imaginary term: alpha_c = spdy*spdx*S/16384, S = checkerboard sum
        {
            float S = 0.0f;
#pragma unroll
            for (int i = 0; i < 8; ++i) S += sWS[i];
            beta += spdy * spdx * (1.0f / 16384.0f) * S;
        }

        // ---- per-wave Ky-real A-fragments from table (ascending pairs) ----
        v2f kyr[16];
#pragma unroll
        for (int kb = 0; kb < 16; ++kb) {
            const int y = w * 16 + ln;
            const int idx0 = (kb * 4 + koff) - y + 127;
            kyr[kb][0] = sHyR[idx0];
            kyr[kb][1] = sHyR[idx0 + 1];
        }

        // ---- two x-halves: p=0 -> x in [0,64), p=1 -> x in [64,128) ----
#pragma unroll
        for (int p = 0; p < 2; ++p) {
            const int xbase = p * 64;
            const int mw = w >> 1;          // stage-1 row tile (u) 0..3
            const int nw = (w & 1) * 2;     // stage-1 col tiles  nw, nw+1

            // Stage 1 (real only): T = P * KxR^T  (M=64 u, K=64 v, N=64 x-half)
            v8f tr0 = (v8f){}, tr1 = (v8f){};
#pragma unroll
            for (int kb = 0; kb < 16; ++kb) {
                v2f a;                       // A = P[u, v] fragment (contiguous)
                const int urow = mw * 16 + ln;
                a[0] = sP[urow * NP + kb * 4 + koff + 0];
                a[1] = sP[urow * NP + kb * 4 + koff + 1];
#pragma unroll
                for (int nt = 0; nt < 2; ++nt) {
                    const int x = xbase + (nw + nt) * 16 + ln;
                    const int idx0 = (kb * 4 + koff) - x + 127;  // ascending pair
                    v2f br;
                    br[0] = sHxR[idx0];
                    br[1] = sHxR[idx0 + 1];
                    if (nt == 0)
                        tr0 = __builtin_amdgcn_wmma_f32_16x16x4_f32(false, a, false, br, (short)0, tr0, false, false);
                    else
                        tr1 = __builtin_amdgcn_wmma_f32_16x16x4_f32(false, a, false, br, (short)0, tr1, false, false);
                }
            }

            __syncthreads();  // previous readers of sTr are done

            // C-layout -> LDS column-major (stride 68): two 16B stores per tile
            {
                const int base = mw * 16 + lhi * 8;
                float* c0 = sTr + ((nw + 0) * 16 + ln) * CST + base;
                float* c1 = sTr + ((nw + 1) * 16 + ln) * CST + base;
#pragma unroll
                for (int m = 0; m < 2; ++m) {
                    v4f q0, q1;
#pragma unroll
                    for (int j = 0; j < 4; ++j) {
                        q0[j] = tr0[4 * m + j];
                        q1[j] = tr1[4 * m + j];
                    }
                    *(v4f*)(c0 + 4 * m) = q0;
                    *(v4f*)(c1 + 4 * m) = q1;
                }
            }

            __syncthreads();  // T ready

            // Stage 2 (real only): D[:, x-half] += KyR * Tr  (K=64 over u)
            // B pair rows (kb*4+koff, +1): contiguous, 8B-aligned -> ds_load_b64
#pragma unroll
            for (int nt = 0; nt < 4; ++nt) {
                v8f d = acc[p * 4 + nt];
                const float* qptr = sTr + (nt * 16 + ln) * CST + koff;
#pragma unroll
                for (int kb = 0; kb < 16; ++kb) {
                    v2f br = *(const v2f*)(qptr + kb * 4);
                    d = __builtin_amdgcn_wmma_f32_16x16x4_f32(false, kyr[kb], false, br, (short)0, d, false, false);
                }
                acc[p * 4 + nt] = d;
            }
        }
        __syncthreads();  // all table/sP/sTr/sWS readers done before next c
    }

    // ---- write canvas with rank-1 checkerboard correction ----
    // out[y,x] = accR[y,x] - beta * (-1)^(y+x)
    float* outb = out + (size_t)b * MP * MP;
#pragma unroll
    for (int nt = 0; nt < 8; ++nt) {
#pragma unroll
        for (int vr = 0; vr < 8; ++vr) {
            const int y = w * 16 + lhi * 8 + vr;
            const int x = nt * 16 + ln;
            const float corr = ((y + x) & 1) ? beta : -beta;
            outb[y * MP + x] = acc[nt][vr] + corr;
        }
    }
}

extern "C" void kernel_launch(void* const* d_in, const int* in_sizes, int n_in,
                              void* d_out, int out_size, void* d_ws, size_t ws_size,
                              hipStream_t stream) {
    const float* patches   = (const float*)d_in[0];  // [128, 64, 64, 16] f32
    const float* positions = (const float*)d_in[1];  // [128, 16, 2]      f32
    float* out = (float*)d_out;                      // [128, 128, 128, 1] f32

    dim3 grid(NB);
    dim3 block(256);
    hipLaunchKernelGGL(reassemble_wmma, grid, block, 0, stream,
                       patches, positions, out);
}